// GELU312_22677427323050
// MI455X (gfx1250) — compile-verified
//
#include <hip/hip_runtime.h>
#include <hip/hip_bf16.h>
#include <math.h>

// CDNA5 / gfx1250 fused kernel: gelu + rank-8 Hebbian gate + cosine-novelty gate.
// Memory-bound: read x (134MB) + write out (134MB) ~= 268MB -> ~11.5us @ 23.3TB/s.
// One block (512 thr, 16 waves) per (b,t) row of D=4096; x & gelu(x) held in
// registers between the stats phase and the finalize phase (V / ema_* stay in L2).
// Intra-wave reductions use V_WMMA_F32_16X16X4_F32, serialized to cap VGPR pressure.

typedef __attribute__((ext_vector_type(2))) float v2f;
typedef __attribute__((ext_vector_type(8))) float v8f;

#define DDIM 4096
#define RANK 8
#define NTHREADS 512
#define NWAVES (NTHREADS / 32)
#define KITER (DDIM / (NTHREADS * 4))   // 2 float4 per thread
#define NQ 11  // t[0..7], sum(out^2), dot(out,ema_dir), sum(ema_dir^2)

__device__ __forceinline__ float gelu_tanh_f(float x) {
    const float c = 0.7978845608028654f;
    float inner = c * (x + 0.044715f * x * x * x);
    return 0.5f * x * (1.0f + tanhf(inner));
}

__device__ __forceinline__ float softplus_f(float v) {
    // stable: max(v,0) + log1p(exp(-|v|))
    return fmaxf(v, 0.0f) + log1pf(expf(-fabsf(v)));
}

// Exact 32-lane sum using WMMA f32 16x16x4.
// A: VGPR0 = per-lane partial, VGPR1 = 0  => sum of all A entries == sum of partials.
// B: all ones (layout-independent). Summing a lane's 8 C/D VGPRs gives the sum of
// A-rows 0-7 (lanes 0-15) or rows 8-15 (lanes 16-31); adding the two halves via
// shfl_xor(16) yields the exact full 32-lane sum in every lane.
__device__ __forceinline__ float wave_reduce_wmma(float partial) {
    v2f a; a.x = partial; a.y = 0.0f;
    v2f b; b.x = 1.0f;    b.y = 1.0f;
    v8f c = {};
    v8f d = __builtin_amdgcn_wmma_f32_16x16x4_f32(
        /*neg_a=*/false, a, /*neg_b=*/false, b,
        /*c_mod=*/(short)0, c, /*reuse_a=*/false, /*reuse_b=*/false);
    float s = ((d[0] + d[1]) + (d[2] + d[3])) + ((d[4] + d[5]) + (d[6] + d[7]));
    s += __shfl_xor(s, 16, 32);
    return s;
}

__global__ __launch_bounds__(NTHREADS) void gelu_gate_fused_kernel(
    const float* __restrict__ x,
    const float* __restrict__ V,           // [D, 8] row-major
    const float* __restrict__ log_tau,
    const float* __restrict__ log_beta_h,
    const float* __restrict__ log_gamma_h,
    const float* __restrict__ ema_power,   // [D]
    const float* __restrict__ ema_dir,     // [D]
    float* __restrict__ out)
{
    const int tid  = threadIdx.x;
    const int lane = tid & 31;
    const int wave = tid >> 5;
    const size_t base = (size_t)blockIdx.x * DDIM;

    const float* xrow = x + base;
    float*       orow = out + base;

    // ---- Phase 1: load x, compute gelu, accumulate per-thread partials ----
    float4 xv[KITER];
    float4 ov[KITER];
    float pt[RANK];
#pragma unroll
    for (int r = 0; r < RANK; ++r) pt[r] = 0.0f;
    float ps2 = 0.0f, psd = 0.0f, pse = 0.0f;

#pragma unroll
    for (int k = 0; k < KITER; ++k) {
        const int idx = tid * 4 + k * (NTHREADS * 4);
        xv[k] = *(const float4*)(xrow + idx);

        float4 o;
        o.x = gelu_tanh_f(xv[k].x);
        o.y = gelu_tanh_f(xv[k].y);
        o.z = gelu_tanh_f(xv[k].z);
        o.w = gelu_tanh_f(xv[k].w);
        ov[k] = o;

        const float4 e = *(const float4*)(ema_dir + idx);
        ps2 += o.x * o.x + o.y * o.y + o.z * o.z + o.w * o.w;
        psd += o.x * e.x + o.y * e.y + o.z * e.z + o.w * e.w;
        pse += e.x * e.x + e.y * e.y + e.z * e.z + e.w * e.w;

        const float xs[4] = { xv[k].x, xv[k].y, xv[k].z, xv[k].w };
#pragma unroll
        for (int j = 0; j < 4; ++j) {
            const size_t dd = (size_t)(idx + j);
            const float4 va = *(const float4*)(V + dd * RANK);
            const float4 vb = *(const float4*)(V + dd * RANK + 4);
            const float xj = xs[j];
            pt[0] = fmaf(xj, va.x, pt[0]);
            pt[1] = fmaf(xj, va.y, pt[1]);
            pt[2] = fmaf(xj, va.z, pt[2]);
            pt[3] = fmaf(xj, va.w, pt[3]);
            pt[4] = fmaf(xj, vb.x, pt[4]);
            pt[5] = fmaf(xj, vb.y, pt[5]);
            pt[6] = fmaf(xj, vb.z, pt[6]);
            pt[7] = fmaf(xj, vb.w, pt[7]);
        }
    }

    // ---- Intra-wave WMMA reductions, serialized to cap live D-tiles ----
    __shared__ __align__(16) float red[NQ][NWAVES];  // transposed for f4 combine

    float parts[NQ];
#pragma unroll
    for (int r = 0; r < RANK; ++r) parts[r] = pt[r];
    parts[8] = ps2; parts[9] = psd; parts[10] = pse;

#pragma unroll
    for (int q = 0; q < NQ; ++q) {
        const float w = wave_reduce_wmma(parts[q]);   // EXEC all-ones here
        if (lane == 0) red[q][wave] = w;
        __builtin_amdgcn_sched_barrier(0);            // don't batch WMMA tiles
    }
    __syncthreads();

    // ---- Cross-wave combine: float4 broadcast reads over the wave axis ----
    float tot[NQ];
#pragma unroll
    for (int q = 0; q < NQ; ++q) {
        const float4* p = (const float4*)(&red[q][0]);
        float s = 0.0f;
#pragma unroll
        for (int w4 = 0; w4 < NWAVES / 4; ++w4) {
            const float4 v4 = p[w4];
            s += (v4.x + v4.y) + (v4.z + v4.w);
        }
        tot[q] = s;
    }

    // ---- Row-level scalars ----
    const float norm_out = sqrtf(tot[8]);
    const float ema_norm = sqrtf(tot[10]);
    float cosv = tot[9] / (fmaxf(norm_out, 1e-12f) * fmaxf(ema_norm, 1e-12f));
    cosv = fminf(fmaxf(cosv, -1.0f), 1.0f);

    const float tau      = expf(log_tau[0]);
    const float gate_cos = expf(-tau * cosv);
    const float beta_h   = softplus_f(log_beta_h[0]);
    const float gamma_h  = softplus_f(log_gamma_h[0]);

    const float t0 = tot[0], t1 = tot[1], t2 = tot[2], t3 = tot[3];
    const float t4 = tot[4], t5 = tot[5], t6 = tot[6], t7 = tot[7];

    // ---- Phase 2: finalize (V and ema_power are L2-resident; x/out in regs) ----
#pragma unroll
    for (int k = 0; k < KITER; ++k) {
        const int idx = tid * 4 + k * (NTHREADS * 4);
        const float4 pw = *(const float4*)(ema_power + idx);
        const float pws[4] = { pw.x, pw.y, pw.z, pw.w };
        const float xs[4]  = { xv[k].x, xv[k].y, xv[k].z, xv[k].w };
        const float os[4]  = { ov[k].x, ov[k].y, ov[k].z, ov[k].w };
        float rs[4];
#pragma unroll
        for (int j = 0; j < 4; ++j) {
            const size_t dd = (size_t)(idx + j);
            const float4 va = *(const float4*)(V + dd * RANK);
            const float4 vb = *(const float4*)(V + dd * RANK + 4);
            float ctx = t0 * va.x + t1 * va.y + t2 * va.z + t3 * va.w
                      + t4 * vb.x + t5 * vb.y + t6 * vb.z + t7 * vb.w;
            const float z  = (xs[j] * ctx) / fmaxf(pws[j], 1e-4f);
            float gh = 1.0f + beta_h * tanhf(gamma_h * z);
            gh = fminf(fmaxf(gh, 0.1f), 5.0f);
            rs[j] = os[j] * gh * gate_cos;
        }
        float4 res;
        res.x = rs[0]; res.y = rs[1]; res.z = rs[2]; res.w = rs[3];
        *(float4*)(orow + idx) = res;
    }
}

extern "C" void kernel_launch(void* const* d_in, const int* in_sizes, int n_in,
                              void* d_out, int out_size, void* d_ws, size_t ws_size,
                              hipStream_t stream) {
    const float* x           = (const float*)d_in[0];
    const float* V           = (const float*)d_in[1];
    const float* log_tau     = (const float*)d_in[2];
    const float* log_beta_h  = (const float*)d_in[3];
    const float* log_gamma_h = (const float*)d_in[4];
    const float* ema_power   = (const float*)d_in[5];
    const float* ema_dir     = (const float*)d_in[6];
    float* out = (float*)d_out;

    const int rows = in_sizes[0] / DDIM;  // B*T = 8192
    hipLaunchKernelGGL(gelu_gate_fused_kernel, dim3(rows), dim3(NTHREADS), 0, stream,
                       x, V, log_tau, log_beta_h, log_gamma_h, ema_power, ema_dir, out);
}